// CustomVoxelization_31774168056339
// MI455X (gfx1250) — compile-verified
//
#include <hip/hip_runtime.h>
#include <stdint.h>

#define BEV_W 848
#define BEV_H 848
#define BEV_HB 10
#define NCELL (BEV_W * BEV_H)          // 719104
#define NCH   (6 + BEV_HB)             // 16
#define BLK   256                      // 8 wave32 per block

// ---- monotonic ordered encoding for float -> uint so global_atomic_max_u32
// implements float max (matches IEEE ordering for non-NaN values).
__device__ __forceinline__ unsigned int f32_to_ordered(float f) {
    unsigned int b = __float_as_uint(f);
    return (b & 0x80000000u) ? ~b : (b | 0x80000000u);
}
__device__ __forceinline__ float ordered_to_f32(unsigned int k) {
    unsigned int b = (k & 0x80000000u) ? (k & 0x7FFFFFFFu) : ~k;
    return __uint_as_float(b);
}

// ---------------------------------------------------------------------------
// Kernel 1: initialize the 16-channel accumulator grid living inside d_out.
// ch0 (zmax) / ch2 (imax) get ordered-encoded 0.0f (0x80000000), rest 0.0f.
// ---------------------------------------------------------------------------
__global__ void bev_init_kernel(unsigned int* __restrict__ feats_u32) {
    int idx = blockIdx.x * blockDim.x + threadIdx.x;
    if (idx >= NCH * NCELL) return;
    int ch = idx / NCELL;
    feats_u32[idx] = (ch == 0 || ch == 2) ? 0x80000000u : 0u;
}

// ---------------------------------------------------------------------------
// Kernel 2: per-point scatter. CDNA5 async DMA: each lane issues
// global_load_async_to_lds_b128 for its own 16B point (VMEM -> LDS, no
// VGPR round-trip, tracked by ASYNCcnt), waits s_wait_asynccnt 0, then
// ds_load_b128's its slice. 6 global atomics into the L2-resident grid.
// ---------------------------------------------------------------------------
__global__ void bev_scatter_kernel(const float* __restrict__ points,
                                   const float* __restrict__ vsize,
                                   const float* __restrict__ crange,
                                   int* __restrict__ coors,
                                   float* __restrict__ feats,
                                   int n) {
    __shared__ __align__(16) float smem[BLK * 4];

    int i = blockIdx.x * blockDim.x + threadIdx.x;
    if (i >= n) return;

    // Prefetch ~8K points ahead (speculative; OOB translation drops silently).
    __builtin_prefetch(points + 4 * (size_t)(i + 8192), 0, 0);

    // ---- CDNA5 async copy: lane's 16B point -> this lane's LDS slot.
    // LDS byte address = low 32 bits of the shared-aperture generic address.
    const uint32_t lds_off =
        (uint32_t)(uintptr_t)(&smem[(size_t)threadIdx.x * 4]);
    const uint64_t gaddr = (uint64_t)(uintptr_t)(points + 4 * (size_t)i);
    asm volatile("global_load_async_to_lds_b128 %0, %1, off"
                 :: "v"(lds_off), "v"(gaddr)
                 : "memory");

    // Uniform params -> scalar loads (overlap with the in-flight DMA).
    const float lx = crange[0], ly = crange[1], lz = crange[2];
    const float vx = vsize[0],  vy = vsize[1],  vz = vsize[2];

    asm volatile("s_wait_asynccnt 0" ::: "memory");
    const float4 p =
        *reinterpret_cast<const float4*>(&smem[(size_t)threadIdx.x * 4]);

    const int cx = (int)floorf((p.x - lx) / vx);
    const int cy = (int)floorf((p.y - ly) / vy);
    const int cz = (int)floorf((p.z - lz) / vz);

    const bool valid = (cx >= 0) & (cx < BEV_W) &
                       (cy >= 0) & (cy < BEV_H) &
                       (cz >= 0) & (cz < 1);

    // coors = valid ? (cz, cy, cx) : (-1,-1,-1)   (row-major N x 3)
    int* cr = coors + 3 * (size_t)i;
    cr[0] = valid ? cz : -1;
    cr[1] = valid ? cy : -1;
    cr[2] = valid ? cx : -1;

    if (!valid) return;  // invalid points write only coors; no grid atomics

    const int cell = cy * BEV_W + cx;
    const float z = p.z, inten = p.w;

    atomicAdd(feats + 4 * NCELL + cell, 1.0f);   // count      -> ch4
    atomicAdd(feats + 1 * NCELL + cell, z);      // z sum      -> ch1
    atomicAdd(feats + 3 * NCELL + cell, inten);  // inten sum  -> ch3

    unsigned int* fu = reinterpret_cast<unsigned int*>(feats);
    atomicMax(fu + 0 * NCELL + cell, f32_to_ordered(z));      // zmax -> ch0
    atomicMax(fu + 2 * NCELL + cell, f32_to_ordered(inten));  // imax -> ch2

    int hb = (int)floorf((z - lz) * 2.0f);       // exact: /0.5 == *2
    hb = min(max(hb, 0), BEV_HB - 1);
    atomicAdd(feats + (6 + hb) * NCELL + cell, 1.0f);
}

// ---------------------------------------------------------------------------
// Kernel 3: per-cell finalize: means, log-count, occupancy, decode maxes.
// ---------------------------------------------------------------------------
__global__ void bev_finalize_kernel(float* __restrict__ feats) {
    int j = blockIdx.x * blockDim.x + threadIdx.x;
    if (j >= NCELL) return;

    unsigned int* fu = reinterpret_cast<unsigned int*>(feats);

    const float cnt = feats[4 * NCELL + j];
    const bool  ne  = cnt >= 1.0f;
    const float cc  = ne ? cnt : 1.0f;

    feats[0 * NCELL + j] = ordered_to_f32(fu[0 * NCELL + j]);  // zmax
    feats[1 * NCELL + j] = feats[1 * NCELL + j] / cc;          // zmean
    feats[2 * NCELL + j] = ordered_to_f32(fu[2 * NCELL + j]);  // imax
    feats[3 * NCELL + j] = feats[3 * NCELL + j] / cc;          // imean
    feats[4 * NCELL + j] = ne ? __logf(cnt + 1.0f) : 0.0f;     // log(count+1)
    feats[5 * NCELL + j] = ne ? 1.0f : 0.0f;                   // occupancy
}

extern "C" void kernel_launch(void* const* d_in, const int* in_sizes, int n_in,
                              void* d_out, int out_size, void* d_ws, size_t ws_size,
                              hipStream_t stream) {
    (void)n_in; (void)out_size; (void)d_ws; (void)ws_size;

    const float* points = (const float*)d_in[0];
    const float* vsize  = (const float*)d_in[1];
    const float* crange = (const float*)d_in[2];

    const int n = in_sizes[0] / 4;               // N points (N,4)

    // Output layout: [ coors int32 (N*3) | feats float32 (16, H, W) ]
    int*   coors = (int*)d_out;
    float* feats = (float*)d_out + 3 * (size_t)n;

    const int init_elems = NCH * NCELL;
    bev_init_kernel<<<(init_elems + BLK - 1) / BLK, BLK, 0, stream>>>(
        (unsigned int*)feats);

    bev_scatter_kernel<<<(n + BLK - 1) / BLK, BLK, 0, stream>>>(
        points, vsize, crange, coors, feats, n);

    bev_finalize_kernel<<<(NCELL + BLK - 1) / BLK, BLK, 0, stream>>>(feats);
}